// AvgPool_22359599743091
// MI455X (gfx1250) — compile-verified
//
#include <hip/hip_runtime.h>

// ---------------------------------------------------------------------------
// Segment mean (sparse avg-pool) for MI455X / gfx1250.
//   d_in[0]: fine_feats  [N_FINE x 64] f32     (512 MB, streamed once, NT)
//   d_in[1]: coarse_ids  [N_FINE]      i32     (8 MB, streamed once, NT)
//   d_out  : [N_COARSE x 64] f32 sums->means   (64 MB, L2-resident accumulator)
//   d_ws   : counts [N_COARSE] f32             (1 MB)
//
// Memory-bound: ~650 MB total traffic -> ~28us floor at 23.3 TB/s; real
// ceiling is L2 atomic throughput on 128M no-return global_atomic_add_f32.
// The 64MB accumulator fits in the 192MB L2; NT hints on the streamed reads
// keep it resident. WMMA/TDM do not apply to a data-dependent scatter.
// ---------------------------------------------------------------------------

typedef float v2f __attribute__((ext_vector_type(2)));
typedef float v4f __attribute__((ext_vector_type(4)));

#define NCHAN 64

// ---- Scatter kernel (emitted first): atomic add of features + counts -------
// Exactly one wave32 per fine row (grid sized to match): lane L holds
// channels {2L, 2L+1} via one coalesced 256B NT float2 load; per-row id load
// is wave-uniform (single L0 transaction). unsafeAtomicAdd lowers to the
// no-return global_atomic_add_f32 hardware atomic (STOREcnt path).
__global__ void __launch_bounds__(256)
ap_scatter_kernel(const float* __restrict__ feats,
                  const int* __restrict__ ids,
                  float* __restrict__ sums,
                  float* __restrict__ cnt,
                  int n_fine) {
    const int lane = threadIdx.x & 31;
    const int row  = (blockIdx.x * blockDim.x + threadIdx.x) >> 5;
    if (row >= n_fine) return;

    // wave-uniform segment id (single request, NT: streamed once)
    const int id = __builtin_nontemporal_load(ids + row);

    // 256B coalesced NT load of this row: lane L -> channels 2L, 2L+1
    const v2f v = __builtin_nontemporal_load(
        (const v2f*)(feats + (size_t)row * NCHAN) + lane);

    float* dst = sums + (size_t)id * NCHAN + (lane << 1);
    unsafeAtomicAdd(dst,     v.x);
    unsafeAtomicAdd(dst + 1, v.y);

    if (lane == 0) unsafeAtomicAdd(cnt + id, 1.0f);
}

// ---- Zero kernel: clear the sum accumulator (d_out) and counts (d_ws) ------
__global__ void __launch_bounds__(256)
ap_zero_kernel(float* __restrict__ out, float* __restrict__ cnt,
               int n_out4, int n_cnt) {
    const int stride = gridDim.x * blockDim.x;
    const v4f z = {0.f, 0.f, 0.f, 0.f};
    v4f* __restrict__ out4 = (v4f*)out;
    for (int k = blockIdx.x * blockDim.x + threadIdx.x; k < n_out4; k += stride)
        out4[k] = z;
    for (int k = blockIdx.x * blockDim.x + threadIdx.x; k < n_cnt; k += stride)
        cnt[k] = 0.f;
}

// ---- Finalize kernel: in-place scale by 1/count ----------------------------
// Empty segments: sums are 0 and inv=0 -> output 0, matching the reference.
// Output is written once and never re-read -> NT store.
__global__ void __launch_bounds__(256)
ap_finalize_kernel(float* __restrict__ out,
                   const float* __restrict__ cnt,
                   int total4) {
    const int stride = gridDim.x * blockDim.x;
    for (int k = blockIdx.x * blockDim.x + threadIdx.x; k < total4; k += stride) {
        const int seg = k >> 4;               // 16 float4's per segment row
        const float c = cnt[seg];
        const float inv = (c > 0.f) ? (1.0f / c) : 0.f;
        v4f v = ((const v4f*)out)[k];
        v.x *= inv; v.y *= inv; v.z *= inv; v.w *= inv;
        __builtin_nontemporal_store(v, (v4f*)out + k);
    }
}

// ---------------------------------------------------------------------------
extern "C" void kernel_launch(void* const* d_in, const int* in_sizes, int n_in,
                              void* d_out, int out_size, void* d_ws, size_t ws_size,
                              hipStream_t stream) {
    const float* feats = (const float*)d_in[0];
    const int*   ids   = (const int*)d_in[1];
    float*       out   = (float*)d_out;
    float*       cnt   = (float*)d_ws;            // n_coarse floats of scratch

    const int n_fine   = in_sizes[1];             // == in_sizes[0] / 64
    const int n_coarse = out_size / NCHAN;

    // 1) zero sums + counts
    {
        const int n_out4 = out_size / 4;
        const int block = 256;
        int grid = (n_out4 + block - 1) / block;
        if (grid > 4096) grid = 4096;             // grid-stride covers the rest
        ap_zero_kernel<<<grid, block, 0, stream>>>(out, cnt, n_out4, n_coarse);
    }

    // 2) scatter: exactly one wave32 per fine row (8 rows per 256-thread block)
    {
        const int block = 256;
        const int rows_per_block = block / 32;
        const int grid = (n_fine + rows_per_block - 1) / rows_per_block;
        ap_scatter_kernel<<<grid, block, 0, stream>>>(feats, ids, out, cnt, n_fine);
    }

    // 3) finalize: divide by counts (multiply by reciprocal), zero-for-empty
    {
        const int total4 = out_size / 4;
        const int block = 256;
        int grid = (total4 + block - 1) / block;
        if (grid > 16384) grid = 16384;
        ap_finalize_kernel<<<grid, block, 0, stream>>>(out, cnt, total4);
    }
}